// GCN_69148973465813
// MI455X (gfx1250) — compile-verified
//
#include <hip/hip_runtime.h>
#include <hip/hip_bf16.h>

typedef __attribute__((ext_vector_type(2))) float v2f;
typedef __attribute__((ext_vector_type(8))) float v8f;

#define H 64
#define WPAD 66   // padded row length for transposed W in LDS (bank spread + 8B align)

// ---------------------------------------------------------------- utilities
__global__ void zero_f32_kernel(float* __restrict__ p, int n) {
    int i = blockIdx.x * blockDim.x + threadIdx.x;
    if (i < n) p[i] = 0.0f;
}

// one thread per edge: accumulate out/in degree (float, hw atomic add)
__global__ void degree_kernel(const int* __restrict__ src, const int* __restrict__ dst,
                              float* __restrict__ out_deg, float* __restrict__ in_deg, int E) {
    int e = blockIdx.x * blockDim.x + threadIdx.x;
    if (e < E) {
        unsafeAtomicAdd(&out_deg[src[e]], 1.0f);
        unsafeAtomicAdd(&in_deg[dst[e]], 1.0f);
    }
}

// in-place: deg -> rsqrt(max(deg,1)) for both arrays
__global__ void norm_kernel(float* __restrict__ norm_out, float* __restrict__ norm_in, int N) {
    int i = blockIdx.x * blockDim.x + threadIdx.x;
    if (i < N) {
        norm_out[i] = rsqrtf(fmaxf(norm_out[i], 1.0f));
        norm_in[i]  = rsqrtf(fmaxf(norm_in[i], 1.0f));
    }
}

// x[i,h] = z_table[z[i], h]
__global__ void embed_kernel(const int* __restrict__ z, const float* __restrict__ z_table,
                             float* __restrict__ x, int N) {
    int t = blockIdx.x * blockDim.x + threadIdx.x;
    if (t < N * H) {
        int i = t >> 6, h = t & (H - 1);
        x[t] = z_table[z[i] * H + h];
    }
}

// SpMM scatter: agg[dst[e],h] += x[src[e],h] * norm_out[src[e]]
// one thread per (edge, feature); consecutive threads cover consecutive h -> coalesced
__global__ void scatter_kernel(const int* __restrict__ src, const int* __restrict__ dst,
                               const float* __restrict__ x, const float* __restrict__ norm_out,
                               float* __restrict__ agg, int E) {
    int t = blockIdx.x * blockDim.x + threadIdx.x;
    if (t < E * H) {
        int e = t >> 6, h = t & (H - 1);
        int s = src[e];
        float v = x[s * H + h] * norm_out[s];
        unsafeAtomicAdd(&agg[dst[e] * H + h], v);
    }
}

// ------------------------------------------------------- fused norm+GEMM+bias(+relu)
// out[m, :] = act( (agg[m, :] * norm_in[m]) @ W + b )
// 256 threads = 8 waves per block; each wave computes a 16-row x 64-col tile
// via V_WMMA_F32_16X16X4_F32 (K-loop of 16 steps, 4 column tiles).
// W is staged TRANSPOSED in LDS so each B fragment (B[kk][n], B[kk+1][n]) is a
// contiguous, 8B-aligned pair -> single ds_load_b64 straight into a WMMA operand.
__global__ __launch_bounds__(256) void gemm_norm_bias_kernel(
    const float* __restrict__ agg, const float* __restrict__ norm_in,
    const float* __restrict__ W, const float* __restrict__ bias,
    float* __restrict__ out, int N, int do_relu) {

    __shared__ float sWt[H * WPAD];   // ~16.9 KB: W^T with padded rows
    for (int i = threadIdx.x; i < H * H; i += 256) {
        int k = i >> 6, n = i & (H - 1);
        sWt[n * WPAD + k] = W[i];     // coalesced read, bank-spread write (stride 66)
    }
    __syncthreads();

    const int wave = threadIdx.x >> 5;
    const int lane = threadIdx.x & 31;
    const int half = lane >> 4;         // 0: lanes 0-15, 1: lanes 16-31
    const int l    = lane & 15;

    const int m0 = (blockIdx.x * 8 + wave) * 16;   // row tile base (wave-uniform)
    if (m0 >= N) return;                            // wave-uniform exit: EXEC stays all-1s

    // A-fragment row for this lane (both halves cover M = m0..m0+15)
    int r = m0 + l;
    if (r >= N - 1) r = N - 1;          // clamp (N % 16 == 0 in practice)
    const float nrm = norm_in[r];
    const float* arow = agg + (size_t)r * H;

    // per-lane base into transposed W: column n = l (+16/32/48 per tile)
    const float* bbase = sWt + l * WPAD;

    v8f acc0 = {}, acc1 = {}, acc2 = {}, acc3 = {};

#pragma unroll
    for (int k0 = 0; k0 < H; k0 += 4) {
        const int kk = k0 + 2 * half;   // lane-half K offset per ISA A/B layout
        v2f a;
        a.x = arow[kk] * nrm;
        a.y = arow[kk + 1] * nrm;

        // contiguous aligned pairs -> ds_load_b64 each
        v2f b0 = *(const v2f*)(bbase + kk);
        v2f b1 = *(const v2f*)(bbase + 16 * WPAD + kk);
        v2f b2 = *(const v2f*)(bbase + 32 * WPAD + kk);
        v2f b3 = *(const v2f*)(bbase + 48 * WPAD + kk);

        acc0 = __builtin_amdgcn_wmma_f32_16x16x4_f32(false, a, false, b0, (short)0, acc0, false, false);
        acc1 = __builtin_amdgcn_wmma_f32_16x16x4_f32(false, a, false, b1, (short)0, acc1, false, false);
        acc2 = __builtin_amdgcn_wmma_f32_16x16x4_f32(false, a, false, b2, (short)0, acc2, false, false);
        acc3 = __builtin_amdgcn_wmma_f32_16x16x4_f32(false, a, false, b3, (short)0, acc3, false, false);
    }

    // D layout: VGPR v holds row M = v + 8*half, col N = l (per 16x16 f32 C/D map)
#pragma unroll
    for (int v = 0; v < 8; ++v) {
        int row = m0 + v + 8 * half;
        if (row < N) {
            float* orow = out + (size_t)row * H;
#pragma unroll
            for (int t = 0; t < 4; ++t) {
                int col = t * 16 + l;
                float val = (t == 0 ? acc0[v] : t == 1 ? acc1[v] : t == 2 ? acc2[v] : acc3[v]);
                val += bias[col];
                if (do_relu) val = fmaxf(val, 0.0f);
                orow[col] = val;
            }
        }
    }
}

// ------------------------------------------------------------------- head MLP
// h = x[p0] * x[p1]; h2 = relu(h @ lin1_w + lin1_b); out = h2 @ lin2_w + lin2_b
__global__ void head_kernel(const float* __restrict__ x, const int* __restrict__ pair,
                            const float* __restrict__ lin1_w, const float* __restrict__ lin1_b,
                            const float* __restrict__ lin2_w, const float* __restrict__ lin2_b,
                            float* __restrict__ out) {
    __shared__ float h[H];
    __shared__ float h2[H];
    const int t = threadIdx.x;
    const int p0 = pair[0], p1 = pair[1];
    h[t] = x[(size_t)p0 * H + t] * x[(size_t)p1 * H + t];
    __syncthreads();
    float s = lin1_b[t];
#pragma unroll 8
    for (int i = 0; i < H; ++i) s += h[i] * lin1_w[i * H + t];
    h2[t] = fmaxf(s, 0.0f);
    __syncthreads();
    if (t == 0) {
        float o = lin2_b[0];
#pragma unroll 8
        for (int j = 0; j < H; ++j) o += h2[j] * lin2_w[j];
        out[0] = o;
    }
}

// ------------------------------------------------------------------- driver
extern "C" void kernel_launch(void* const* d_in, const int* in_sizes, int n_in,
                              void* d_out, int out_size, void* d_ws, size_t ws_size,
                              hipStream_t stream) {
    const int*   z       = (const int*)  d_in[0];
    const int*   src     = (const int*)  d_in[1];
    const int*   dst     = (const int*)  d_in[2];
    const int*   pair    = (const int*)  d_in[3];
    const float* z_table = (const float*)d_in[4];
    const float* W1      = (const float*)d_in[5];
    const float* b1      = (const float*)d_in[6];
    const float* W2      = (const float*)d_in[7];
    const float* b2      = (const float*)d_in[8];
    const float* W3      = (const float*)d_in[9];
    const float* b3      = (const float*)d_in[10];
    const float* lin1_w  = (const float*)d_in[11];
    const float* lin1_b  = (const float*)d_in[12];
    const float* lin2_w  = (const float*)d_in[13];
    const float* lin2_b  = (const float*)d_in[14];
    float* out = (float*)d_out;

    const int N = in_sizes[0];
    const int E = in_sizes[1];

    // workspace layout
    float* norm_out = (float*)d_ws;          // N
    float* norm_in  = norm_out + N;          // N
    float* bufA     = norm_in + N;           // N*H  (x)
    float* bufB     = bufA + (size_t)N * H;  // N*H  (agg)

    const int TPB = 256;
    auto cdiv = [](long a, long b) { return (int)((a + b - 1) / b); };

    // degrees + norms
    zero_f32_kernel<<<cdiv(2L * N, TPB), TPB, 0, stream>>>(norm_out, 2 * N);
    degree_kernel<<<cdiv(E, TPB), TPB, 0, stream>>>(src, dst, norm_out, norm_in, E);
    norm_kernel<<<cdiv(N, TPB), TPB, 0, stream>>>(norm_out, norm_in, N);

    // embedding
    embed_kernel<<<cdiv((long)N * H, TPB), TPB, 0, stream>>>(z, z_table, bufA, N);

    const int gemm_blocks = cdiv(N, 128);  // 8 waves/block * 16 rows/wave

    // layer 1 (relu)
    zero_f32_kernel<<<cdiv((long)N * H, TPB), TPB, 0, stream>>>(bufB, N * H);
    scatter_kernel<<<cdiv((long)E * H, TPB), TPB, 0, stream>>>(src, dst, bufA, norm_out, bufB, E);
    gemm_norm_bias_kernel<<<gemm_blocks, TPB, 0, stream>>>(bufB, norm_in, W1, b1, bufA, N, 1);

    // layer 2 (relu)
    zero_f32_kernel<<<cdiv((long)N * H, TPB), TPB, 0, stream>>>(bufB, N * H);
    scatter_kernel<<<cdiv((long)E * H, TPB), TPB, 0, stream>>>(src, dst, bufA, norm_out, bufB, E);
    gemm_norm_bias_kernel<<<gemm_blocks, TPB, 0, stream>>>(bufB, norm_in, W2, b2, bufA, N, 1);

    // layer 3 (no relu)
    zero_f32_kernel<<<cdiv((long)N * H, TPB), TPB, 0, stream>>>(bufB, N * H);
    scatter_kernel<<<cdiv((long)E * H, TPB), TPB, 0, stream>>>(src, dst, bufA, norm_out, bufB, E);
    gemm_norm_bias_kernel<<<gemm_blocks, TPB, 0, stream>>>(bufB, norm_in, W3, b3, bufA, N, 0);

    // head
    head_kernel<<<1, H, 0, stream>>>(bufA, pair, lin1_w, lin1_b, lin2_w, lin2_b, out);
}